// MambaGatedConvSSM3D_19628000543405
// MI455X (gfx1250) — compile-verified
//
#include <hip/hip_runtime.h>
#include <hip/hip_bf16.h>
#include <math.h>

// ---------------------------------------------------------------------------
// MambaGatedConvSSM3D for MI455X (gfx1250, wave32)
//   B=2, T=16, H=W=C=64
// Pipeline (all on `stream`):
//   K1 freq_response : A_f/B_f = DFT of circularly-padded 3x3x3 kernels
//   K2 delta GEMM    : (131072x64)@(64x64) via V_WMMA_F32_16X16X4_F32 + softplus
//   K3 fft W (fwd)   : real->complex, 64-pt FFT per thread, fully in registers
//   K4 fft H (fwd)   : in-place complex
//   K5 fft T (fwd)   : in-place complex (16-pt)
//   K6 scan over T   : h_t = a_t*h_{t-1} + b_t (complex), in-place into Xf
//   K7 fft T (inv), K8 fft H (inv), K9 fft W (inv) -> real out * 1/65536
// ---------------------------------------------------------------------------

typedef __attribute__((ext_vector_type(2))) float v2f;
typedef __attribute__((ext_vector_type(8))) float v8f;

#define FPI 3.14159265358979f

__device__ __forceinline__ float sigmoidf_(float v) {
  return 1.0f / (1.0f + __expf(-v));
}
__device__ __forceinline__ float softplusf_(float v) {
  // logaddexp(v, 0) = max(v,0) + log1p(exp(-|v|))
  return fmaxf(v, 0.0f) + log1pf(__expf(-fabsf(v)));
}

template <int LOGN>
__device__ __forceinline__ int brev_(int x) {
  int r = 0;
#pragma unroll
  for (int i = 0; i < LOGN; ++i) { r = (r << 1) | (x & 1); x >>= 1; }
  return r;
}

// Fully unrolled iterative radix-2 FFT in registers. After full unrolling all
// indices and twiddle angles are compile-time constants -> SROA keeps re/im
// in VGPRs (no scratch).
template <int N, int LOGN, bool INV>
__device__ __forceinline__ void fft_inreg(float (&re)[N], float (&im)[N]) {
#pragma unroll
  for (int i = 0; i < N; ++i) {
    const int j = brev_<LOGN>(i);
    if (j > i) {
      float t;
      t = re[i]; re[i] = re[j]; re[j] = t;
      t = im[i]; im[i] = im[j]; im[j] = t;
    }
  }
#pragma unroll
  for (int s = 1; s <= LOGN; ++s) {
    const int m = 1 << s;
    const int mh = m >> 1;
#pragma unroll
    for (int j = 0; j < mh; ++j) {
      const float ang = (INV ? 2.0f : -2.0f) * FPI * (float)j / (float)m;
      const float wr = __cosf(ang);
      const float wi = __sinf(ang);
#pragma unroll
      for (int k = j; k < N; k += m) {
        const int i1 = k + mh;
        const float tr = wr * re[i1] - wi * im[i1];
        const float ti = wr * im[i1] + wi * re[i1];
        re[i1] = re[k] - tr; im[i1] = im[k] - ti;
        re[k] += tr;         im[k] += ti;
      }
    }
  }
}

// ---------------------------------------------------------------------------
// K1: frequency response of 3x3x3 kernels. Output layout (t,h,w,c), c inner.
// All phases are exact multiples of 2*pi/64 (t/16 == 4t/64) -> integer mod 64
// gives exact range reduction before __cosf/__sinf.
// ---------------------------------------------------------------------------
__global__ __launch_bounds__(256) void freq_response_kernel(
    const float* __restrict__ Ak, const float* __restrict__ Bk,
    float2* __restrict__ Af, float2* __restrict__ Bf) {
  const int idx = blockIdx.x * 256 + threadIdx.x;   // (t,h,w,c), 4,194,304
  const float* kin = (blockIdx.y == 0) ? Ak : Bk;
  float2* outp     = (blockIdx.y == 0) ? Af : Bf;
  const int c = idx & 63;
  const int w = (idx >> 6) & 63;
  const int h = (idx >> 12) & 63;
  const int t = idx >> 18;
  float accr = 0.0f, acci = 0.0f;
#pragma unroll
  for (int i = 0; i < 3; ++i) {
    const int pt = (i - 1) & 15;     // (i-1) mod 16
#pragma unroll
    for (int j = 0; j < 3; ++j) {
      const int ph = (j - 1) & 63;   // (j-1) mod 64
#pragma unroll
      for (int l = 0; l < 3; ++l) {
        const int pw = (l - 1) & 63;
        const float kv = kin[c * 27 + (i * 3 + j) * 3 + l];
        const int phase = (4 * t * pt + h * ph + w * pw) & 63;
        const float ang = -(2.0f * FPI / 64.0f) * (float)phase;
        accr = fmaf(kv, __cosf(ang), accr);
        acci = fmaf(kv, __sinf(ang), acci);
      }
    }
  }
  outp[idx] = make_float2(accr, acci);
}

// ---------------------------------------------------------------------------
// K2: delta = softplus(x @ delta_W + delta_b) with V_WMMA_F32_16X16X4_F32.
// One wave computes a 16x64 output block (4 accumulator tiles, K=64 in 16
// steps of 4).
//
// delta_W is staged in LDS in row-pair-interleaved form:
//     lWp[(k/2)*WPAIR + n*2 + (k&1)] = W[k][n]
// so the B operand {W[k][n], W[k+1][n]} for the wave32 4x16 layout is ONE
// aligned ds_load_b64 into an even VGPR pair (no VALU shuffles). WPAIR=160
// (not 128) pads row-pairs so lanes 0-15 (rows K,K+1) and lanes 16-31 (rows
// K+2,K+3) land on disjoint halves of the 64 LDS banks: delta = 160 mod 64 =
// 32 -> conflict-free 32-lane x 8B access.
// ---------------------------------------------------------------------------
#define WPAIR 160

__global__ __launch_bounds__(128) void delta_gemm_softplus_kernel(
    const float* __restrict__ x, const float* __restrict__ Wm,
    const float* __restrict__ bias, float* __restrict__ out) {
  __shared__ float lWp[32 * WPAIR];
  for (int i = threadIdx.x; i < 64 * 64; i += 128) {
    const int k = i >> 6;
    const int n = i & 63;
    lWp[(k >> 1) * WPAIR + n * 2 + (k & 1)] = Wm[i];
  }
  __syncthreads();

  const int lane = threadIdx.x & 31;
  const int wave = threadIdx.x >> 5;
  const size_t rowBase = ((size_t)blockIdx.x * 4 + wave) * 16;
  const int m  = lane & 15;          // row (A) / col (B,C/D) within tile
  const int kk = (lane >> 4) << 1;   // lanes 16-31 hold K+2/K+3
  const float* aRow = x + (rowBase + (size_t)m) * 64;
  __builtin_prefetch(aRow, 0, 1);    // -> global_prefetch_b8

  v8f acc0 = {}, acc1 = {}, acc2 = {}, acc3 = {};
#pragma unroll
  for (int k0 = 0; k0 < 64; k0 += 4) {
    const v2f a = *(const v2f*)(aRow + k0 + kk);          // A[m][k0+kk .. +1]
    const float* wrp = &lWp[((k0 + kk) >> 1) * WPAIR + m * 2];
    const v2f b0 = *(const v2f*)(wrp + 0);                // n0 = 0
    const v2f b1 = *(const v2f*)(wrp + 32);               // n0 = 16
    const v2f b2 = *(const v2f*)(wrp + 64);               // n0 = 32
    const v2f b3 = *(const v2f*)(wrp + 96);               // n0 = 48
    acc0 = __builtin_amdgcn_wmma_f32_16x16x4_f32(false, a, false, b0, (short)0, acc0, false, false);
    acc1 = __builtin_amdgcn_wmma_f32_16x16x4_f32(false, a, false, b1, (short)0, acc1, false, false);
    acc2 = __builtin_amdgcn_wmma_f32_16x16x4_f32(false, a, false, b2, (short)0, acc2, false, false);
    acc3 = __builtin_amdgcn_wmma_f32_16x16x4_f32(false, a, false, b3, (short)0, acc3, false, false);
  }

  // C/D layout: vgpr r -> M = r (lanes<16) or 8+r (lanes>=16); N = lane&15.
  const size_t mOff = (size_t)(lane >> 4) * 8;
  const float bv0 = bias[m], bv1 = bias[16 + m], bv2 = bias[32 + m], bv3 = bias[48 + m];
#pragma unroll
  for (int r = 0; r < 8; ++r) {
    float* orow = out + (rowBase + mOff + (size_t)r) * 64;
    orow[m]      = softplusf_(acc0[r] + bv0);
    orow[16 + m] = softplusf_(acc1[r] + bv1);
    orow[32 + m] = softplusf_(acc2[r] + bv2);
    orow[48 + m] = softplusf_(acc3[r] + bv3);
  }
}

// ---------------------------------------------------------------------------
// FFT passes. x layout: [((b*16+t)*64 + h)*64 + w]*64 + c   (c innermost).
// Consecutive threads take consecutive c -> every strided line access is a
// fully coalesced 128B/256B request.
// ---------------------------------------------------------------------------
__global__ __launch_bounds__(256) void fft_w_fwd_kernel(
    const float* __restrict__ x, float2* __restrict__ Xf) {
  const int idx = blockIdx.x * 256 + threadIdx.x;   // (b,t,h,c), 131072
  const int c = idx & 63;
  const int bth = idx >> 6;
  const float* src = x + (size_t)bth * 4096 + c;
  float re[64], im[64];
#pragma unroll
  for (int w = 0; w < 64; ++w) { re[w] = src[(size_t)w * 64]; im[w] = 0.0f; }
  fft_inreg<64, 6, false>(re, im);
  float2* dst = Xf + (size_t)bth * 4096 + c;
#pragma unroll
  for (int w = 0; w < 64; ++w) dst[(size_t)w * 64] = make_float2(re[w], im[w]);
}

template <bool INV>
__global__ __launch_bounds__(256) void fft_h_pass_kernel(float2* __restrict__ Xf) {
  const int idx = blockIdx.x * 256 + threadIdx.x;   // (b,t,w,c), 131072
  const int c = idx & 63;
  const int w = (idx >> 6) & 63;
  const int bt = idx >> 12;
  float2* p = Xf + (size_t)bt * 262144 + (size_t)w * 64 + c;
  float re[64], im[64];
#pragma unroll
  for (int h = 0; h < 64; ++h) {
    const float2 v = p[(size_t)h * 4096]; re[h] = v.x; im[h] = v.y;
  }
  fft_inreg<64, 6, INV>(re, im);
#pragma unroll
  for (int h = 0; h < 64; ++h) p[(size_t)h * 4096] = make_float2(re[h], im[h]);
}

template <bool INV>
__global__ __launch_bounds__(256) void fft_t_pass_kernel(float2* __restrict__ Xf) {
  const int idx = blockIdx.x * 256 + threadIdx.x;   // (b,h,w,c), 524288
  const int hwc = idx & 0x3FFFF;                    // h*4096 + w*64 + c
  const int b = idx >> 18;
  float2* p = Xf + (size_t)b * 4194304 + hwc;
  float re[16], im[16];
#pragma unroll
  for (int t = 0; t < 16; ++t) {
    const float2 v = p[(size_t)t * 262144]; re[t] = v.x; im[t] = v.y;
  }
  fft_inreg<16, 4, INV>(re, im);
#pragma unroll
  for (int t = 0; t < 16; ++t) p[(size_t)t * 262144] = make_float2(re[t], im[t]);
}

__global__ __launch_bounds__(256) void fft_w_inv_real_kernel(
    const float2* __restrict__ Xf, float* __restrict__ out) {
  const int idx = blockIdx.x * 256 + threadIdx.x;   // (b,t,h,c), 131072
  const int c = idx & 63;
  const int bth = idx >> 6;
  const float2* src = Xf + (size_t)bth * 4096 + c;
  float re[64], im[64];
#pragma unroll
  for (int w = 0; w < 64; ++w) {
    const float2 v = src[(size_t)w * 64]; re[w] = v.x; im[w] = v.y;
  }
  fft_inreg<64, 6, true>(re, im);
  float* dst = out + (size_t)bth * 4096 + c;
  const float scale = 1.0f / 65536.0f;              // 1/(T*H*W)
#pragma unroll
  for (int w = 0; w < 64; ++w) dst[(size_t)w * 64] = re[w] * scale;
}

// ---------------------------------------------------------------------------
// K6: gated complex recurrence over T, in-place on Xf (h_t overwrites x_f_t).
//   a_t = sigmoid(fb + fs*x) * A_f          (complex)
//   b_t = sigmoid(ib + is*x) * delta * B_f * x_f
//   h_t = a_t * h_{t-1} + b_t,  h_{-1} = 0
// ---------------------------------------------------------------------------
__global__ __launch_bounds__(256) void scan_time_kernel(
    const float* __restrict__ x, const float* __restrict__ delta,
    const float2* __restrict__ Af, const float2* __restrict__ Bf,
    const float* __restrict__ fb, const float* __restrict__ fs,
    const float* __restrict__ ib, const float* __restrict__ isc,
    float2* __restrict__ Hf) {
  const int idx = blockIdx.x * 256 + threadIdx.x;   // (b,h,w,c), 524288
  const int c = idx & 63;
  const int hwc = idx & 0x3FFFF;
  const int b = idx >> 18;
  const float fbv = fb[c], fsv = fs[c], ibv = ib[c], isv = isc[c];
  float hr = 0.0f, hi = 0.0f;
#pragma unroll
  for (int t = 0; t < 16; ++t) {
    const size_t sp = (size_t)(b * 16 + t) * 262144 + hwc;
    const size_t fq = (size_t)t * 262144 + hwc;
    const float xv = x[sp];
    const float fgate = sigmoidf_(fbv + fsv * xv);
    const float gg = sigmoidf_(ibv + isv * xv) * delta[sp];
    const float2 Av = Af[fq];
    const float2 Bv = Bf[fq];
    const float2 Xv = Hf[sp];
    const float ar = fgate * Av.x, ai = fgate * Av.y;
    const float br = gg * (Bv.x * Xv.x - Bv.y * Xv.y);
    const float bi = gg * (Bv.x * Xv.y + Bv.y * Xv.x);
    const float nr = ar * hr - ai * hi + br;
    const float ni = ar * hi + ai * hr + bi;
    hr = nr; hi = ni;
    Hf[sp] = make_float2(hr, hi);
  }
}

// ---------------------------------------------------------------------------
// Launch. setup_inputs order:
//   0:x 1:A_kernel 2:B_kernel 3:forget_bias 4:forget_scale 5:input_bias
//   6:input_scale 7:delta_W 8:delta_b
// Workspace: Xf (8,388,608 float2) | Af (4,194,304 float2) | Bf (4,194,304
// float2) | delta (8,388,608 float) = 160 MiB.
// ---------------------------------------------------------------------------
extern "C" void kernel_launch(void* const* d_in, const int* in_sizes, int n_in,
                              void* d_out, int out_size, void* d_ws, size_t ws_size,
                              hipStream_t stream) {
  (void)in_sizes; (void)n_in; (void)out_size; (void)ws_size;
  const float* x   = (const float*)d_in[0];
  const float* Ak  = (const float*)d_in[1];
  const float* Bk  = (const float*)d_in[2];
  const float* fb  = (const float*)d_in[3];
  const float* fs  = (const float*)d_in[4];
  const float* ib  = (const float*)d_in[5];
  const float* isc = (const float*)d_in[6];
  const float* dW  = (const float*)d_in[7];
  const float* db  = (const float*)d_in[8];
  float* out = (float*)d_out;

  float2* Xf    = (float2*)d_ws;
  float2* Af    = Xf + 8388608;
  float2* Bf    = Af + 4194304;
  float*  delta = (float*)(Bf + 4194304);

  // K1: A_f / B_f (grid.y selects kernel); 4,194,304 threads each.
  freq_response_kernel<<<dim3(16384, 2), 256, 0, stream>>>(Ak, Bk, Af, Bf);
  // K2: delta GEMM via WMMA. 131072 rows / (4 waves * 16 rows) = 2048 blocks.
  delta_gemm_softplus_kernel<<<2048, 128, 0, stream>>>(x, dW, db, delta);
  // K3-K5: forward 3D FFT of x.
  fft_w_fwd_kernel<<<512, 256, 0, stream>>>(x, Xf);
  fft_h_pass_kernel<false><<<512, 256, 0, stream>>>(Xf);
  fft_t_pass_kernel<false><<<2048, 256, 0, stream>>>(Xf);
  // K6: gated recurrence over T (in-place -> Xf holds h_f).
  scan_time_kernel<<<2048, 256, 0, stream>>>(x, delta, Af, Bf, fb, fs, ib, isc, Xf);
  // K7-K9: inverse 3D FFT, real output scaled by 1/65536.
  fft_t_pass_kernel<true><<<2048, 256, 0, stream>>>(Xf);
  fft_h_pass_kernel<true><<<512, 256, 0, stream>>>(Xf);
  fft_w_inv_real_kernel<<<512, 256, 0, stream>>>(Xf, out);
}